// Grapher_42580305773110
// MI455X (gfx1250) — compile-verified
//
#include <hip/hip_runtime.h>
#include <hip/hip_bf16.h>

// DGCNN edge-conv block for MI455X (gfx1250, wave32, WMMA).
// Shapes: B=8, C=96, H=W=56 (N=3136), K=9, C2=192.
// All matmuls via V_WMMA_F32_16X16X4_F32 (full f32 precision -> kNN indices
// match the f32 reference; compute is tiny vs. 23.3 TB/s HBM anyway).
//
// Workspace layout (bytes):
//   feat : 0          .. 9,633,792   (B,N,C)  point-major, post-BN features
//   fn   : 9,633,792  .. 19,267,584 (B,N,C)  L2-normalized features
//   sq   : 19,267,584 .. 19,367,936 (B,N)    sum(fn^2) per point
//   nn   : 19,367,936 .. 20,271,104 (B,N,K)  int32 neighbor indices
//   agg  : 20,271,104 .. 39,538,688 (B,C2,N) channel-major edge-max output
// Total ~39.6 MB.

#define NPIX 3136
#define NB   8
#define CIN  96
#define C2CH 192
#define KNN  9
#define NT16 196   // NPIX/16

typedef __attribute__((ext_vector_type(2))) float v2f;
typedef __attribute__((ext_vector_type(8))) float v8f;

__device__ __forceinline__ v8f wmma_f32x4(v2f a, v2f b, v8f c) {
  // D = A(16x4) * B(4x16) + C(16x16), all f32.  8-arg form:
  // (neg_a, A, neg_b, B, c_mod, C, reuse_a, reuse_b)
  return __builtin_amdgcn_wmma_f32_16x16x4_f32(false, a, false, b, (short)0, c,
                                               false, false);
}

__device__ __forceinline__ v8f v8f_zero() {
  v8f z;
#pragma unroll
  for (int i = 0; i < 8; ++i) z[i] = 0.0f;
  return z;
}

// ---------------------------------------------------------------------------
// Kernel 1: h = BN(W1 @ x + b1); write feat (point-major), fn (normalized),
// sq (sum fn^2).  One wave per 16-pixel tile, 6 row-tiles (all 96 out chans)
// so the per-point channel norm stays in-wave (one shfl_xor across hi halves).
// ---------------------------------------------------------------------------
__global__ __launch_bounds__(256) void dgcnn_conv1(
    const float* __restrict__ x, const float* __restrict__ W1,
    const float* __restrict__ b1, const float* __restrict__ g1,
    const float* __restrict__ be1, const float* __restrict__ m1,
    const float* __restrict__ v1, float* __restrict__ feat,
    float* __restrict__ fn, float* __restrict__ sq) {
  __shared__ float lW[CIN * CIN];       // 36.9 KB
  __shared__ float lscale[CIN], lshift[CIN];
  const int t = threadIdx.x;
  for (int i = t; i < CIN * CIN; i += 256) lW[i] = W1[i];
  if (t < CIN) {
    float inv = g1[t] * rsqrtf(v1[t] + 1e-5f);
    lscale[t] = inv;
    lshift[t] = b1[t] * inv + be1[t] - m1[t] * inv;
  }
  __syncthreads();

  const int wave = t >> 5, lane = t & 31;
  const int col = lane & 15, hi = lane >> 4;
  const int tile = blockIdx.x * 8 + wave;         // 0..1567
  const int b = tile / NT16;
  const int n0 = (tile % NT16) * 16;
  const int n = n0 + col;
  const float* xb = x + (size_t)b * CIN * NPIX;

  v8f acc[6];
#pragma unroll
  for (int rt = 0; rt < 6; ++rt) acc[rt] = v8f_zero();

#pragma unroll 1
  for (int kk = 0; kk < CIN; kk += 4) {
    const int kb = kk + 2 * hi;
    v2f bm;
    bm.x = xb[(size_t)kb * NPIX + n];
    bm.y = xb[(size_t)(kb + 1) * NPIX + n];
#pragma unroll
    for (int rt = 0; rt < 6; ++rt) {
      const int row = rt * 16 + col;
      v2f am = *(const v2f*)&lW[row * CIN + kb];   // kb even -> 8B aligned
      acc[rt] = wmma_f32x4(am, bm, acc[rt]);
    }
  }

  // Epilogue: BN fold, feat write, per-point norm, fn/sq write.
  float vals[48];
  float ss = 0.0f;
  float* fp = feat + ((size_t)b * NPIX + n) * CIN;
#pragma unroll
  for (int rt = 0; rt < 6; ++rt) {
#pragma unroll
    for (int r = 0; r < 8; ++r) {
      const int o = rt * 16 + r + 8 * hi;
      float v = acc[rt][r] * lscale[o] + lshift[o];
      vals[rt * 8 + r] = v;
      ss += v * v;
      fp[o] = v;
    }
  }
  ss += __shfl_xor(ss, 16, 32);                    // combine hi halves
  float rinv = 1.0f / fmaxf(sqrtf(ss), 1e-12f);
  float* fq = fn + ((size_t)b * NPIX + n) * CIN;
#pragma unroll
  for (int rt = 0; rt < 6; ++rt) {
#pragma unroll
    for (int r = 0; r < 8; ++r) {
      const int o = rt * 16 + r + 8 * hi;
      fq[o] = vals[rt * 8 + r] * rinv;
    }
  }
  if (lane < 16) sq[(size_t)b * NPIX + n] = ss * rinv * rinv;
}

// ---------------------------------------------------------------------------
// Kernel 2: Gram G = Fn * Fn^T (per batch) fused with top-9 selection.
// One wave owns 16 *columns* m (B-operand preloaded: 48 VGPRs) and streams
// 196 row tiles.  Ranking key = 2*G[n,m] - sq[n]  (== -dist + const per m).
// Each lane keeps a top-9 for its column over its 8-row subsets; the two hi
// halves merge via LDS with (key desc, index asc) stable ordering.
// ---------------------------------------------------------------------------
__global__ __launch_bounds__(32) void dgcnn_knn(
    const float* __restrict__ fn, const float* __restrict__ sq,
    int* __restrict__ nn_idx) {
  __shared__ float lv[16][18];
  __shared__ int li[16][18];
  const int lane = threadIdx.x & 31;
  const int col = lane & 15, hi = lane >> 4;
  const int b = blockIdx.x / NT16;
  const int m0 = (blockIdx.x % NT16) * 16;
  const int m = m0 + col;
  const float* fb = fn + (size_t)b * NPIX * CIN;
  const float* sb = sq + (size_t)b * NPIX;

  v2f Bop[24];
#pragma unroll
  for (int s = 0; s < 24; ++s)
    Bop[s] = *(const v2f*)&fb[(size_t)m * CIN + 4 * s + 2 * hi];

  float bestv[KNN];
  int besti[KNN];
#pragma unroll
  for (int i = 0; i < KNN; ++i) { bestv[i] = -1e30f; besti[i] = 0; }

#pragma unroll 1
  for (int nt = 0; nt < NT16; ++nt) {
    const int n0 = nt * 16;
    v8f acc = v8f_zero();
    const float* fr = fb + (size_t)(n0 + col) * CIN + 2 * hi;
#pragma unroll
    for (int s = 0; s < 24; ++s) {
      v2f A = *(const v2f*)&fr[4 * s];
      acc = wmma_f32x4(A, Bop[s], acc);
    }
    // Scan the 8 rows this lane holds for column m.
#pragma unroll
    for (int r = 0; r < 8; ++r) {
      const int nrow = n0 + r + 8 * hi;
      const float key = 2.0f * acc[r] - sb[nrow];
      if (key > bestv[KNN - 1]) {
        float v = key; int id = nrow;
#pragma unroll
        for (int j = 0; j < KNN; ++j) {
          if (v > bestv[j]) {       // strict: equal keys keep earlier index
            float tv = bestv[j]; int ti = besti[j];
            bestv[j] = v; besti[j] = id;
            v = tv; id = ti;
          }
        }
      }
    }
  }

  // Merge hi-half lists (each sorted desc, indices asc on ties).
#pragma unroll
  for (int i = 0; i < KNN; ++i) {
    lv[col][hi * KNN + i] = bestv[i];
    li[col][hi * KNN + i] = besti[i];
  }
  __syncthreads();
  if (lane < 16) {
    int ia = 0, ib = KNN;
    int* outp = nn_idx + ((size_t)b * NPIX + m) * KNN;
#pragma unroll
    for (int k = 0; k < KNN; ++k) {
      bool takeA;
      if (ia >= KNN) takeA = false;
      else if (ib >= 2 * KNN) takeA = true;
      else {
        float va = lv[col][ia], vb = lv[col][ib];
        takeA = (va > vb) || (va == vb && li[col][ia] < li[col][ib]);
      }
      outp[k] = takeA ? li[col][ia++] : li[col][ib++];
    }
  }
}

// ---------------------------------------------------------------------------
// Kernel 3: edge MLP + max over K.
// agg[o,n] = relu( max_k (Wg[:, :96]@x_i + Wg[:, 96:]@(x_j-x_i)) + bg[o] )
// (bias+ReLU hoisted out of the k-max: both are monotone/shift-invariant).
// Block = 128 thr (4 waves) per 16-point tile; x_i and all 9 (x_j-x_i)
// blocks staged in LDS (~63 KB); each wave keeps its 3 row-tiles of Wg in
// 96 VGPRs (Wg stays L2-resident: 147 KB vs 192 MB L2).
// ---------------------------------------------------------------------------
__global__ __launch_bounds__(128) void dgcnn_edge(
    const float* __restrict__ feat, const int* __restrict__ nn_idx,
    const float* __restrict__ Wg, const float* __restrict__ bg,
    float* __restrict__ agg) {
  __shared__ float e0[CIN * 16];            // x_i   [c][p]   6144 B
  __shared__ float ej[KNN][CIN * 16];       // x_j-x_i per k  55296 B
  __shared__ int lidx[16 * KNN];
  __shared__ float lbg[C2CH];
  const int t = threadIdx.x;
  const int b = blockIdx.x / NT16;
  const int n0 = (blockIdx.x % NT16) * 16;
  const float* fb = feat + (size_t)b * NPIX * CIN;

  if (t < 16 * KNN) lidx[t] = nn_idx[((size_t)b * NPIX + n0) * KNN + t];
  for (int i = t; i < C2CH; i += 128) lbg[i] = bg[i];
  {
    const int p = t & 15, c0 = t >> 4;      // c0 in 0..7
    const float* fi = fb + (size_t)(n0 + p) * CIN;
    for (int c = c0; c < CIN; c += 8) e0[c * 16 + p] = fi[c];
  }
  __syncthreads();
  {
    const int p = t & 15, c0 = t >> 4;
    for (int k = 0; k < KNN; ++k) {
      const int j = lidx[p * KNN + k];
      const float* fj = fb + (size_t)j * CIN;
      for (int c = c0; c < CIN; c += 8)
        ej[k][c * 16 + p] = fj[c] - e0[c * 16 + p];
    }
  }
  __syncthreads();

  const int lane = t & 31, wave = t >> 5;
  const int col = lane & 15, hi = lane >> 4;

#pragma unroll 1
  for (int rr = 0; rr < 3; ++rr) {
    const int rt = wave * 3 + rr;            // 12 row tiles over 4 waves
    const int row = rt * 16 + col;
    v2f Aop[48];                             // Wg row operands (96 VGPRs)
    const float* wr = Wg + (size_t)row * C2CH + 2 * hi;
#pragma unroll
    for (int s = 0; s < 48; ++s) Aop[s] = *(const v2f*)&wr[4 * s];

    v8f mx;
#pragma unroll 1
    for (int k = 0; k < KNN; ++k) {
      v8f acc = v8f_zero();
#pragma unroll
      for (int s = 0; s < 24; ++s) {         // channels 0..95: x_i
        const int kb = 4 * s + 2 * hi;
        v2f Bm;
        Bm.x = e0[kb * 16 + col];
        Bm.y = e0[(kb + 1) * 16 + col];
        acc = wmma_f32x4(Aop[s], Bm, acc);
      }
      const float* ejk = &ej[k][0];
#pragma unroll
      for (int s = 0; s < 24; ++s) {         // channels 96..191: x_j - x_i
        const int kb = 4 * s + 2 * hi;
        v2f Bm;
        Bm.x = ejk[kb * 16 + col];
        Bm.y = ejk[(kb + 1) * 16 + col];
        acc = wmma_f32x4(Aop[24 + s], Bm, acc);
      }
      if (k == 0) {
        mx = acc;
      } else {
#pragma unroll
        for (int r = 0; r < 8; ++r) mx[r] = fmaxf(mx[r], acc[r]);
      }
    }
#pragma unroll
    for (int r = 0; r < 8; ++r) {
      const int o = rt * 16 + r + 8 * hi;
      agg[((size_t)b * C2CH + o) * NPIX + n0 + col] =
          fmaxf(mx[r] + lbg[o], 0.0f);
    }
  }
}

// ---------------------------------------------------------------------------
// Kernel 4: out = BN(W2 @ agg + b2) + x.  rt-outer so W2 row operands live in
// 96 VGPRs (no >64KB LDS needed); agg is channel-major -> coalesced B loads.
// ---------------------------------------------------------------------------
__global__ __launch_bounds__(256) void dgcnn_out(
    const float* __restrict__ agg, const float* __restrict__ W2,
    const float* __restrict__ b2, const float* __restrict__ g2,
    const float* __restrict__ be2, const float* __restrict__ m2,
    const float* __restrict__ v2, const float* __restrict__ x,
    float* __restrict__ out) {
  __shared__ float lscale[CIN], lshift[CIN];
  const int t = threadIdx.x;
  if (t < CIN) {
    float inv = g2[t] * rsqrtf(v2[t] + 1e-5f);
    lscale[t] = inv;
    lshift[t] = b2[t] * inv + be2[t] - m2[t] * inv;
  }
  __syncthreads();

  const int wave = t >> 5, lane = t & 31;
  const int col = lane & 15, hi = lane >> 4;
  const int tile = blockIdx.x * 8 + wave;
  const int b = tile / NT16;
  const int n0 = (tile % NT16) * 16;
  const int n = n0 + col;
  const float* ab = agg + (size_t)b * C2CH * NPIX;
  const float* xb = x + (size_t)b * CIN * NPIX;

#pragma unroll 1
  for (int rt = 0; rt < 6; ++rt) {
    const int row = rt * 16 + col;
    v2f Aop[48];
    const float* wr = W2 + (size_t)row * C2CH + 2 * hi;
#pragma unroll
    for (int s = 0; s < 48; ++s) Aop[s] = *(const v2f*)&wr[4 * s];

    v8f acc = v8f_zero();
#pragma unroll
    for (int s = 0; s < 48; ++s) {
      const int kb = 4 * s + 2 * hi;
      v2f Bm;
      Bm.x = ab[(size_t)kb * NPIX + n];
      Bm.y = ab[(size_t)(kb + 1) * NPIX + n];
      acc = wmma_f32x4(Aop[s], Bm, acc);
    }
#pragma unroll
    for (int r = 0; r < 8; ++r) {
      const int o = rt * 16 + r + 8 * hi;
      out[((size_t)b * CIN + o) * NPIX + n] =
          acc[r] * lscale[o] + lshift[o] + xb[(size_t)o * NPIX + n];
    }
  }
}

// ---------------------------------------------------------------------------
extern "C" void kernel_launch(void* const* d_in, const int* in_sizes, int n_in,
                              void* d_out, int out_size, void* d_ws,
                              size_t ws_size, hipStream_t stream) {
  (void)in_sizes; (void)n_in; (void)out_size; (void)ws_size;
  const float* x   = (const float*)d_in[0];
  const float* W1  = (const float*)d_in[1];
  const float* b1  = (const float*)d_in[2];
  const float* g1  = (const float*)d_in[3];
  const float* be1 = (const float*)d_in[4];
  const float* m1  = (const float*)d_in[5];
  const float* v1  = (const float*)d_in[6];
  const float* Wg  = (const float*)d_in[7];
  const float* bg  = (const float*)d_in[8];
  const float* W2  = (const float*)d_in[9];
  const float* b2  = (const float*)d_in[10];
  const float* g2  = (const float*)d_in[11];
  const float* be2 = (const float*)d_in[12];
  const float* m2  = (const float*)d_in[13];
  const float* v2  = (const float*)d_in[14];
  float* out = (float*)d_out;

  char* ws = (char*)d_ws;
  float* feat = (float*)(ws + 0);
  float* fn   = (float*)(ws + 9633792);
  float* sq   = (float*)(ws + 19267584);
  int*   nn   = (int*)  (ws + 19367936);
  float* agg  = (float*)(ws + 20271104);   // total ws use ~39.6 MB

  dgcnn_conv1<<<196, 256, 0, stream>>>(x, W1, b1, g1, be1, m1, v1, feat, fn, sq);
  dgcnn_knn<<<NB * NT16, 32, 0, stream>>>(fn, sq, nn);
  dgcnn_edge<<<NB * NT16, 128, 0, stream>>>(feat, nn, Wg, bg, agg);
  dgcnn_out<<<196, 256, 0, stream>>>(agg, W2, b2, g2, be2, m2, v2, x, out);
}